// FastLearnableTimeWarping_3281355014676
// MI455X (gfx1250) — compile-verified
//
#include <hip/hip_runtime.h>
#include <hip/hip_bf16.h>

typedef __attribute__((ext_vector_type(2))) float v2f;
typedef __attribute__((ext_vector_type(8))) float v8f;

#define B_   64
#define T1_  320
#define T2_  320
#define D_   768
#define WD_  32
#define BW_  81          // max(T1,T2)/4 + 1
#define NEGV (-1.0e30f)

// ---------------------------------------------------------------------------
// Kernel 1: projection  xw = (x @ W^T + b) * scale + shift,  plus row norms.
// One wave per row (lane = output channel, WD==32==wave32). float4 K-loop.
// ---------------------------------------------------------------------------
__global__ void proj_kernel(const float* __restrict__ X,
                            const float* __restrict__ W,
                            const float* __restrict__ bias,
                            const float* __restrict__ scale,
                            const float* __restrict__ shift,
                            float* __restrict__ XW,
                            float* __restrict__ XN,
                            int nrows) {
    int wave = threadIdx.x >> 5;
    int lane = threadIdx.x & 31;
    int row  = blockIdx.x * 8 + wave;
    if (row >= nrows) return;
    const float4* xr = reinterpret_cast<const float4*>(X + (size_t)row * D_);
    const float4* wr = reinterpret_cast<const float4*>(W + (size_t)lane * D_);
    float acc = 0.f;
#pragma unroll 4
    for (int k = 0; k < D_ / 4; ++k) {
        float4 a = xr[k];
        float4 w = wr[k];
        acc += a.x * w.x + a.y * w.y + a.z * w.z + a.w * w.w;
    }
    float v = (acc + bias[lane]) * scale[lane] + shift[lane];
    XW[(size_t)row * WD_ + lane] = v;
    float n = v * v;
#pragma unroll
    for (int off = 16; off; off >>= 1) n += __shfl_xor(n, off, 32);
    if (lane == 0) XN[row] = n;
}

// ---------------------------------------------------------------------------
// Kernel 2: dist[b,i,j] = |xw_i|^2 + |yw_j|^2 - 2 * xw_i . yw_j
// One wave per 16x16 (i,j) tile, K=32 via 8x V_WMMA_F32_16X16X4_F32.
// ---------------------------------------------------------------------------
__global__ void dist_wmma_kernel(const float* __restrict__ XW,
                                 const float* __restrict__ YW,
                                 const float* __restrict__ XN,
                                 const float* __restrict__ YN,
                                 float* __restrict__ DIST) {
    int b    = blockIdx.y;
    int wave = threadIdx.x >> 5;
    int lane = threadIdx.x & 31;
    int half = lane >> 4;       // 0: lanes 0-15, 1: lanes 16-31
    int l16  = lane & 15;
    int tile = blockIdx.x * 8 + wave;            // 0..399
    int ti = tile / (T2_ / 16);
    int tj = tile % (T2_ / 16);
    int i0 = ti * 16, j0 = tj * 16;

    const float* arow = XW + ((size_t)b * T1_ + i0 + l16) * WD_;  // A: M = l16
    const float* brow = YW + ((size_t)b * T2_ + j0 + l16) * WD_;  // B: N = l16

    v8f c = {};
#pragma unroll
    for (int k0 = 0; k0 < WD_; k0 += 4) {
        v2f a, bb;
        a.x  = arow[k0 + 2 * half];
        a.y  = arow[k0 + 2 * half + 1];
        bb.x = brow[k0 + 2 * half];
        bb.y = brow[k0 + 2 * half + 1];
        c = __builtin_amdgcn_wmma_f32_16x16x4_f32(false, a, false, bb,
                                                  (short)0, c, false, false);
    }
    // C/D layout: VGPR v -> M = v + 8*half, N = l16
    float yn = YN[(size_t)b * T2_ + j0 + l16];
#pragma unroll
    for (int v = 0; v < 8; ++v) {
        int i = i0 + v + 8 * half;
        int j = j0 + l16;
        float d = XN[(size_t)b * T1_ + i] + yn - 2.0f * c[v];
        DIST[((size_t)b * T1_ + i) * T2_ + j] = d;
    }
}

// ---------------------------------------------------------------------------
// Kernel 3: banded soft-DTW DP (gamma = 1), anti-diagonal wavefront.
// One block per batch, 320 threads (thread j == column t2), 3 rotating
// LDS diagonals, one barrier per diagonal. Out-of-band cells = 0 (and read).
// ---------------------------------------------------------------------------
__global__ void dtw_dp_kernel(const float* __restrict__ DIST,
                              float* __restrict__ P) {
    __shared__ float buf[3][T2_];
    int b = blockIdx.x;
    int j = threadIdx.x;                 // t2
    const float* db = DIST + (size_t)b * T1_ * T2_;
    float* pb = P + (size_t)b * T1_ * T2_;

    for (int d = 0; d < T1_ + T2_ - 1; ++d) {
        int t1 = d - j;
        int cur = d % 3, p1 = (d + 2) % 3, p2 = (d + 1) % 3;
        float val = 0.f;
        bool active = (t1 >= 0) && (t1 < T1_);
        if (active) {
            float dd = db[(size_t)t1 * T2_ + j];
            float up = buf[p1][j];                       // (t1-1, j)
            float lf = (j > 0) ? buf[p1][j - 1] : 0.f;   // (t1, j-1)
            float dg = (j > 0) ? buf[p2][j - 1] : 0.f;   // (t1-1, j-1)
            float z0 = (t1 > 0 && j > 0) ? -dg : NEGV;
            float z1 = (t1 > 0) ? -up : NEGV;
            float z2 = (j > 0) ? -lf : NEGV;
            float m = fmaxf(z0, fmaxf(z1, z2));
            float softmin = 0.f;
            if (t1 > 0 || j > 0)
                softmin = -(m + __logf(__expf(z0 - m) + __expf(z1 - m) + __expf(z2 - m)));
            val = dd + softmin;
            bool in_band = (j >= t1 - BW_) && (j <= t1 + BW_ - 1);
            val = in_band ? val : 0.f;
            pb[(size_t)t1 * T2_ + j] = val;
        }
        if (active) buf[cur][j] = val;
        __syncthreads();
    }
}

// ---------------------------------------------------------------------------
// Kernel 4: in-place row softmax of -P  (one block per (b,t1) row).
// ---------------------------------------------------------------------------
__global__ void softmax_kernel(float* __restrict__ P) {
    __shared__ float red[16];
    __shared__ float bval;
    int j = threadIdx.x;
    int wave = j >> 5, lane = j & 31;
    float* row = P + (size_t)blockIdx.x * T2_;
    float z = -row[j];
    float m = z;
#pragma unroll
    for (int off = 16; off; off >>= 1) m = fmaxf(m, __shfl_xor(m, off, 32));
    if (lane == 0) red[wave] = m;
    __syncthreads();
    if (j == 0) {
        float mm = red[0];
        for (int w = 1; w < T2_ / 32; ++w) mm = fmaxf(mm, red[w]);
        bval = mm;
    }
    __syncthreads();
    float M = bval;
    float e = __expf(z - M);
    float s = e;
#pragma unroll
    for (int off = 16; off; off >>= 1) s += __shfl_xor(s, off, 32);
    __syncthreads();
    if (lane == 0) red[wave] = s;
    __syncthreads();
    if (j == 0) {
        float ss = 0.f;
        for (int w = 0; w < T2_ / 32; ++w) ss += red[w];
        bval = ss;
    }
    __syncthreads();
    row[j] = e / bval;
}

// ---------------------------------------------------------------------------
// Kernel 5: inv column-sum  invcs[b,t2] = 1 / (sum_t1 path[b,t1,t2] + 1e-10)
// ---------------------------------------------------------------------------
__global__ void colsum_kernel(const float* __restrict__ P,
                              float* __restrict__ INVCS) {
    int idx = blockIdx.x * blockDim.x + threadIdx.x;   // 0..B*T2-1
    if (idx >= B_ * T2_) return;
    int b = idx / T2_, t2 = idx % T2_;
    const float* pb = P + (size_t)b * T1_ * T2_ + t2;
    float acc = 0.f;
    for (int t1 = 0; t1 < T1_; ++t1) acc += pb[(size_t)t1 * T2_];
    INVCS[idx] = 1.0f / (acc + 1e-10f);
}

// ---------------------------------------------------------------------------
// Kernel 6: aligned[b,s,d] = (1/cs[b,s]) * sum_t path[b,t,s] * x[b,t,d]
// One wave per 16x16 output tile; K=320 via 80x V_WMMA_F32_16X16X4_F32.
// Two independent accumulation chains (even/odd K-slices) so the matrix
// pipe never stalls on the D->C dependency; 2x unroll = 4 WMMA + 16 loads
// in flight per iteration.
// ---------------------------------------------------------------------------
__global__ void aligned_wmma_kernel(const float* __restrict__ PATH,
                                    const float* __restrict__ X,
                                    const float* __restrict__ INVCS,
                                    float* __restrict__ OUT) {
    int b    = blockIdx.y;
    int wave = threadIdx.x >> 5;
    int lane = threadIdx.x & 31;
    int half = lane >> 4;
    int l16  = lane & 15;
    int tile = blockIdx.x * 8 + wave;            // 0..959
    int ts = tile / (D_ / 16);                   // s-tile 0..19
    int td = tile % (D_ / 16);                   // d-tile 0..47
    int s0 = ts * 16, d0 = td * 16;

    const float* pb = PATH + (size_t)b * T1_ * T2_ + s0 + l16;   // A col base
    const float* xb = X + (size_t)b * T1_ * D_ + d0 + l16;       // B col base

    v8f c0 = {};
    v8f c1 = {};
#pragma unroll 2
    for (int t0 = 0; t0 < T1_; t0 += 8) {
        int ka = t0 + 2 * half;
        v2f a0, b0, a1, b1;
        // A[s][t] = path[t][s] : M = l16 -> s0+l16, K = ka (+1)
        a0.x = pb[(size_t)ka * T2_];
        a0.y = pb[(size_t)(ka + 1) * T2_];
        b0.x = xb[(size_t)ka * D_];
        b0.y = xb[(size_t)(ka + 1) * D_];
        a1.x = pb[(size_t)(ka + 4) * T2_];
        a1.y = pb[(size_t)(ka + 5) * T2_];
        b1.x = xb[(size_t)(ka + 4) * D_];
        b1.y = xb[(size_t)(ka + 5) * D_];
        c0 = __builtin_amdgcn_wmma_f32_16x16x4_f32(false, a0, false, b0,
                                                   (short)0, c0, false, false);
        c1 = __builtin_amdgcn_wmma_f32_16x16x4_f32(false, a1, false, b1,
                                                   (short)0, c1, false, false);
    }
#pragma unroll
    for (int v = 0; v < 8; ++v) {
        int s = s0 + v + 8 * half;
        int d = d0 + l16;
        float o = (c0[v] + c1[v]) * INVCS[(size_t)b * T2_ + s];
        OUT[((size_t)b * T2_ + s) * D_ + d] = o;
    }
}

// ---------------------------------------------------------------------------
extern "C" void kernel_launch(void* const* d_in, const int* in_sizes, int n_in,
                              void* d_out, int out_size, void* d_ws, size_t ws_size,
                              hipStream_t stream) {
    const float* x      = (const float*)d_in[0];  // [B,T1,D]
    const float* y      = (const float*)d_in[1];  // [B,T2,D]
    const float* W      = (const float*)d_in[2];  // [WD,D]
    const float* bias   = (const float*)d_in[3];  // [WD]
    const float* wscale = (const float*)d_in[4];  // [WD]
    const float* wshift = (const float*)d_in[5];  // [WD]

    float* out     = (float*)d_out;
    float* aligned = out;                                   // [B,T2,D]
    float* path    = out + (size_t)B_ * T2_ * D_;           // [B,T1,T2]
    float* dist    = aligned;  // reuse aligned region as dist scratch (dead until k6)

    float* ws    = (float*)d_ws;
    float* XW    = ws;                                      // B*T1*WD
    float* YW    = XW + (size_t)B_ * T1_ * WD_;             // B*T2*WD
    float* XN    = YW + (size_t)B_ * T2_ * WD_;             // B*T1
    float* YN    = XN + (size_t)B_ * T1_;                   // B*T2
    float* INVCS = YN + (size_t)B_ * T2_;                   // B*T2

    const int nrows = B_ * T1_;

    // 1) projections
    proj_kernel<<<nrows / 8, 256, 0, stream>>>(x, W, bias, wscale, wshift, XW, XN, nrows);
    proj_kernel<<<nrows / 8, 256, 0, stream>>>(y, W, bias, wscale, wshift, YW, YN, nrows);

    // 2) pairwise squared distances (WMMA f32)
    dim3 gdist((T1_ / 16) * (T2_ / 16) / 8, B_);
    dist_wmma_kernel<<<gdist, 256, 0, stream>>>(XW, YW, XN, YN, dist);

    // 3) banded soft-DTW wavefront DP -> path region of d_out
    dtw_dp_kernel<<<B_, T2_, 0, stream>>>(dist, path);

    // 4) in-place row softmax of -P
    softmax_kernel<<<B_ * T1_, T2_, 0, stream>>>(path);

    // 5) inverse column sums
    colsum_kernel<<<(B_ * T2_ + 255) / 256, 256, 0, stream>>>(path, INVCS);

    // 6) aligned = (path^T @ x) / colsum (WMMA f32)  — overwrites dist scratch
    dim3 gal((T2_ / 16) * (D_ / 16) / 8, B_);
    aligned_wmma_kernel<<<gal, 256, 0, stream>>>(path, x, INVCS, aligned);
}